// CorrelationAlign_29884382445645
// MI455X (gfx1250) — compile-verified
//
#include <hip/hip_runtime.h>
#include <stdint.h>

// Problem constants (from reference): b=16, h=w=32, H=W=63.
constexpr int KB      = 16;
constexpr int KH      = 32;
constexpr int KW      = 32;
constexpr int KHO     = 63;          // 2h-1
constexpr int KWO     = 63;          // 2w-1
constexpr int KHW     = KH * KW;     // 1024
constexpr int KOPLANE = KHO * KWO;   // 3969

typedef float v4f __attribute__((ext_vector_type(4)));

// One workgroup per (b, a, i), taken straight from the 3D grid (no div/mod).
// p = a-31+i is block-uniform. Stage IN[b][p][q][i][j] (32x32 tile over q,j)
// into LDS via async DMA, then stream 63 coalesced 128B output rows
// (c = 0..62), reading the LDS diagonal tile[c-31+j][j] and masking invalid q.
__global__ __launch_bounds__(256) void
CorrelationAlign_kernel(const float* __restrict__ in, float* __restrict__ out) {
  __shared__ float tile[KH * KW];   // tile[q][j]; 16B-aligned rows of 128B

  const int i = blockIdx.x;               // 0..31
  const int a = blockIdx.y;               // 0..62
  const int b = blockIdx.z;               // 0..15
  const int p = a - (KH - 1) + i;
  const bool pvalid = (p >= 0) && (p < KH);   // block-uniform

  const int tid = threadIdx.x;

  if (pvalid) {
    // Thread t DMAs the 16B chunk (q = t/8, j = 4*(t%8)..+3): global -> LDS.
    const int q  = tid >> 3;
    const int jv = (tid & 7) << 2;
    const float* gsrc =
        in + ((size_t)((b * KHW + p * KW + q) * KH + i) * KW + jv);
    const uint32_t lds_addr = (uint32_t)(size_t)&tile[q * KW + jv];
    asm volatile("global_load_async_to_lds_b128 %0, %1, off"
                 :: "v"(lds_addr), "v"(gsrc)
                 : "memory");
#if __has_builtin(__builtin_amdgcn_s_wait_asynccnt)
    __builtin_amdgcn_s_wait_asynccnt(0);
#else
    asm volatile("s_wait_asynccnt 0" ::: "memory");
#endif
  }
  __syncthreads();

  // Output slice base for (b, a, :, i, :): row c is 128B contiguous.
  float* obase = out + (size_t)(b * KOPLANE + a * KWO) * KHW + i * KW;
  for (int idx = tid; idx < KWO * (KW / 4); idx += 256) {  // 504 float4 chunks
    const int c  = idx >> 3;
    const int jv = (idx & 7) << 2;
    v4f v = {0.f, 0.f, 0.f, 0.f};
    if (pvalid) {
#pragma unroll
      for (int k = 0; k < 4; ++k) {
        const int j = jv + k;
        const int q = c - (KW - 1) + j;
        if (q >= 0 && q < KW) v[k] = tile[q * KW + j];
      }
    }
    // Streamed output: written once, never re-read by this kernel.
    __builtin_nontemporal_store(v, (v4f*)(obase + (size_t)c * KHW + jv));
  }
}

extern "C" void kernel_launch(void* const* d_in, const int* in_sizes, int n_in,
                              void* d_out, int out_size, void* d_ws, size_t ws_size,
                              hipStream_t stream) {
  (void)in_sizes; (void)n_in; (void)out_size; (void)d_ws; (void)ws_size;
  const float* in  = (const float*)d_in[0];
  float*       out = (float*)d_out;
  dim3 grid(KH, KHO, KB);   // (i, a, b) = (32, 63, 16) -> 32256 blocks
  CorrelationAlign_kernel<<<grid, 256, 0, stream>>>(in, out);
}